// GNNModel_2241972928666
// MI455X (gfx1250) — compile-verified
//
#include <hip/hip_runtime.h>

// GCN 2-layer, 2 graphs, fp32 end-to-end using V_WMMA_F32_16X16X4_F32 for the
// dense stages and hardware fp32 global atomics for the edge scatters.

typedef __attribute__((ext_vector_type(2))) float v2f;
typedef __attribute__((ext_vector_type(8))) float v8f;

#define IN_FEATS 128
#define HID      64
#define NC       32

// ---------------- utility kernels ----------------

__global__ void fill_zero_kernel(float* __restrict__ p, int n) {
    int i = blockIdx.x * blockDim.x + threadIdx.x;
    if (i < n) p[i] = 0.0f;
}

__global__ void deg_kernel(const int* __restrict__ src, const int* __restrict__ dst,
                           float* __restrict__ deg_out, float* __restrict__ deg_in, int E) {
    int e = blockIdx.x * blockDim.x + threadIdx.x;
    if (e < E) {
        unsafeAtomicAdd(&deg_out[src[e]], 1.0f);
        unsafeAtomicAdd(&deg_in[dst[e]], 1.0f);
    }
}

__global__ void rsqrt_clip_kernel(float* __restrict__ d, int n) {
    int i = blockIdx.x * blockDim.x + threadIdx.x;
    if (i < n) d[i] = rsqrtf(fmaxf(d[i], 1.0f));
}

// agg1 = relu(agg1 * rs_in[row] + b[col])   (in place), F = HID
__global__ void bias_relu_kernel(float* __restrict__ a, const float* __restrict__ rs_in,
                                 const float* __restrict__ b, int n) {
    int i = blockIdx.x * blockDim.x + threadIdx.x;
    if (i < n * HID) {
        int row = i >> 6;          // /64
        int col = i & (HID - 1);
        float v = a[i] * rs_in[row] + b[col];
        a[i] = fmaxf(v, 0.0f);
    }
}

// out += 0.5 * (agg2 * rs_in[row] + b[col]),  F = NC
__global__ void out_acc_kernel(float* __restrict__ out, const float* __restrict__ a,
                               const float* __restrict__ rs_in, const float* __restrict__ b,
                               int n) {
    int i = blockIdx.x * blockDim.x + threadIdx.x;
    if (i < n * NC) {
        int row = i >> 5;          // /32
        int col = i & (NC - 1);
        out[i] += 0.5f * (a[i] * rs_in[row] + b[col]);
    }
}

// ---------------- edge scatter: agg[dst] += h[src], F floats/row ----------------
// One thread per (edge, 4-feature chunk): coalesced float4 gather + 4 hw atomics.
template <int F>
__global__ void scatter_kernel(const float* __restrict__ h, const int* __restrict__ src,
                               const int* __restrict__ dst, float* __restrict__ agg, int E) {
    constexpr int C = F / 4;
    int idx = blockIdx.x * blockDim.x + threadIdx.x;
    if (idx >= E * C) return;
    int e = idx / C;
    int c = idx - e * C;
    int s = src[e];
    int d = dst[e];
    const float4 v = *(const float4*)&h[(long)s * F + c * 4];
    float* p = &agg[(long)d * F + c * 4];
    unsafeAtomicAdd(p + 0, v.x);
    unsafeAtomicAdd(p + 1, v.y);
    unsafeAtomicAdd(p + 2, v.z);
    unsafeAtomicAdd(p + 3, v.w);
}

// ---------------- WMMA fp32 GEMM:  H[n,OUT] = (X[n,IN] * rs[row]) @ W[IN,OUT] ----------------
// 256 threads = 8 waves; wave w of block b owns rows [ (b*8+w)*16 , +16 ).
// W is transposed into padded LDS so each B-fragment is one contiguous 8B read.
template <int IN, int OUT>
__global__ void __launch_bounds__(256)
gemm_gc_kernel(const float* __restrict__ X, const float* __restrict__ rs,
               const float* __restrict__ W, float* __restrict__ H, int n) {
    constexpr int LDW = IN + 4;                 // pad to spread LDS banks
    __shared__ float lw[OUT * LDW];

    // cooperative transpose W[IN][OUT] -> lw[col][k]
    for (int i = threadIdx.x; i < IN * OUT; i += 256) {
        int k = i / OUT;
        int c = i - k * OUT;
        lw[c * LDW + k] = W[i];
    }
    __syncthreads();

    const int lane = threadIdx.x & 31;
    const int wave = threadIdx.x >> 5;
    const int r0 = (blockIdx.x * 8 + wave) * 16;
    if (r0 >= n) return;                        // wave-uniform: EXEC stays all-ones

    const int mrow = r0 + (lane & 15);
    const int koff = (lane >> 4) * 2;           // K offset for upper half-wave (A and B)
    const float rsv = rs[mrow];
    const float* xrow = X + (long)mrow * IN;
    const int col = lane & 15;

    constexpr int CT = OUT / 16;
    v8f acc[CT] = {};

    for (int k = 0; k < IN; k += 4) {
        // A fragment: A[m][k+koff], A[m][k+koff+1]  (contiguous global float2)
        float2 av = *(const float2*)(xrow + k + koff);
        v2f a;
        a.x = av.x * rsv;
        a.y = av.y * rsv;
#pragma unroll
        for (int c = 0; c < CT; ++c) {
            // B fragment: W[k+koff][col], W[k+koff+1][col]  (contiguous in transposed LDS)
            float2 bv = *(const float2*)&lw[(c * 16 + col) * LDW + k + koff];
            v2f b;
            b.x = bv.x;
            b.y = bv.y;
            acc[c] = __builtin_amdgcn_wmma_f32_16x16x4_f32(
                false, a, false, b, (short)0, acc[c], false, false);
        }
    }

    // D layout: lane 0-15 -> N=lane, VGPR r -> M=r ; lanes 16-31 -> M=r+8
    const int mhi = (lane >> 4) * 8;
#pragma unroll
    for (int c = 0; c < CT; ++c) {
#pragma unroll
        for (int r = 0; r < 8; ++r) {
            H[(long)(r0 + mhi + r) * OUT + c * 16 + col] = acc[c][r];
        }
    }
}

// ---------------- launcher ----------------

extern "C" void kernel_launch(void* const* d_in, const int* in_sizes, int n_in,
                              void* d_out, int out_size, void* d_ws, size_t ws_size,
                              hipStream_t stream) {
    const float* x    = (const float*)d_in[0];
    const int*   src1 = (const int*)d_in[1];
    const int*   dst1 = (const int*)d_in[2];
    const int*   src2 = (const int*)d_in[3];
    const int*   dst2 = (const int*)d_in[4];
    const float* W1   = (const float*)d_in[5];
    const float* b1   = (const float*)d_in[6];
    const float* W2   = (const float*)d_in[7];
    const float* b2   = (const float*)d_in[8];
    float* out = (float*)d_out;

    const int N = in_sizes[0] / IN_FEATS;   // 100000
    const int E = in_sizes[1];              // 1600000

    // workspace carve-up (floats)
    float* ws   = (float*)d_ws;
    float* rsO  = ws;                       // N   (deg_out -> rsqrt)
    float* rsI  = rsO + N;                  // N   (deg_in  -> rsqrt)
    float* h    = rsI + N;                  // N*HID
    float* agg1 = h + (size_t)N * HID;      // N*HID  (becomes h1 in place)
    float* h2   = agg1 + (size_t)N * HID;   // N*NC
    float* agg2 = h2 + (size_t)N * NC;      // N*NC

    const int T = 256;
    auto cdiv = [](long a, long b) { return (int)((a + b - 1) / b); };

    const int tiles   = N / 16;                 // N % 16 == 0
    const int gblocks = cdiv(tiles, 8);

    // out = 0
    fill_zero_kernel<<<cdiv((long)N * NC, T), T, 0, stream>>>(out, N * NC);

    for (int g = 0; g < 2; ++g) {
        const int* src = (g == 0) ? src1 : src2;
        const int* dst = (g == 0) ? dst1 : dst2;

        // degrees -> rsqrt(clip(deg,1))
        fill_zero_kernel<<<cdiv(N, T), T, 0, stream>>>(rsO, N);
        fill_zero_kernel<<<cdiv(N, T), T, 0, stream>>>(rsI, N);
        deg_kernel<<<cdiv(E, T), T, 0, stream>>>(src, dst, rsO, rsI, E);
        rsqrt_clip_kernel<<<cdiv(N, T), T, 0, stream>>>(rsO, N);
        rsqrt_clip_kernel<<<cdiv(N, T), T, 0, stream>>>(rsI, N);

        // layer 1: h = (x * rsO) @ W1 ; agg1 = scatter ; h1 = relu(agg1*rsI + b1)
        gemm_gc_kernel<IN_FEATS, HID><<<gblocks, T, 0, stream>>>(x, rsO, W1, h, N);
        fill_zero_kernel<<<cdiv((long)N * HID, T), T, 0, stream>>>(agg1, N * HID);
        scatter_kernel<HID><<<cdiv((long)E * (HID / 4), T), T, 0, stream>>>(h, src, dst, agg1, E);
        bias_relu_kernel<<<cdiv((long)N * HID, T), T, 0, stream>>>(agg1, rsI, b1, N);

        // layer 2: h2 = (h1 * rsO) @ W2 ; agg2 = scatter ; out += 0.5*(agg2*rsI + b2)
        gemm_gc_kernel<HID, NC><<<gblocks, T, 0, stream>>>(agg1, rsO, W2, h2, N);
        fill_zero_kernel<<<cdiv((long)N * NC, T), T, 0, stream>>>(agg2, N * NC);
        scatter_kernel<NC><<<cdiv((long)E * (NC / 4), T), T, 0, stream>>>(h2, src, dst, agg2, E);
        out_acc_kernel<<<cdiv((long)N * NC, T), T, 0, stream>>>(out, agg2, rsI, b2, N);
    }
}